// DWAMiddleLayer_77498389889523
// MI455X (gfx1250) — compile-verified
//
#include <hip/hip_runtime.h>

typedef __attribute__((ext_vector_type(2))) float v2f;
typedef __attribute__((ext_vector_type(8))) float v8f;

#define DWA_B   2048
#define DWA_N   512
#define DWA_DA  256
#define DWA_DB  256
#define DWA_R   4
#define DWA_D   4096
#define DWA_NR  (DWA_N * DWA_R)              // 2048
#define DWA_UEND (DWA_DB * DWA_R)            // 1024
#define DWA_VEND (DWA_UEND + DWA_R * DWA_DA) // 2048

__device__ __forceinline__ v8f wmma_f32(v2f a, v2f b, v8f c) {
  // D = A(16x4, f32) * B(4x16, f32) + C(16x16, f32)
  return __builtin_amdgcn_wmma_f32_16x16x4_f32(
      /*neg_a=*/false, a, /*neg_b=*/false, b,
      /*c_mod=*/(short)0, c, /*reuse_a=*/false, /*reuse_b=*/false);
}

// CDNA5 async global->LDS copy (ASYNCcnt-tracked, bypasses VGPRs).
// LDS operand: flat-LDS aperture keeps the logical LDS address in addr[31:0].
__device__ __forceinline__ void async_copy_b128(void* lds_dst, const void* gsrc) {
  const uint32_t lds_off = (uint32_t)(uintptr_t)lds_dst;
  asm volatile("global_load_async_to_lds_b128 %0, %1, off"
               :: "v"(lds_off), "v"(gsrc) : "memory");
}
__device__ __forceinline__ void wait_async0() {
  asm volatile("s_wait_asynccnt 0x0" ::: "memory");
}

// Kernel 1: S[b, j] = alpha[b, j>>2] * sum_a h_A[b,a] * V[j>>2, j&3, a]
// GEMM M=2048 (b), N=2048 (j = n*4+r), K=256 (a).
// Workgroup tile: 16 rows x 128 cols, 8 waves, one 16x16 WMMA tile per wave.
__global__ __launch_bounds__(256) void k_tv(
    const float* __restrict__ hA, const float* __restrict__ pool,
    const float* __restrict__ alpha, float* __restrict__ S)
{
  __shared__ float As[16][260]; // 16x256 A tile, padded
  const int rowbase = blockIdx.x * 16;
  const int colbase = blockIdx.y * 128;

  // Async-stage the 16x256 h_A tile into LDS (16B per lane per iteration).
  #pragma unroll
  for (int i = threadIdx.x; i < 16 * 64; i += 256) {
    const int r  = i >> 6;
    const int c4 = (i & 63) << 2;
    async_copy_b128(&As[r][c4], hA + (size_t)(rowbase + r) * DWA_DA + c4);
  }

  const int wave = threadIdx.x >> 5;
  const int lane = threadIdx.x & 31;
  const int ln   = lane & 15;          // N index within tile / M index for A reads
  const int kh   = (lane >> 4) << 1;   // K sub-offset: 0 (lanes 0-15) or 2 (16-31)
  const int j    = colbase + wave * 16 + ln;
  const int n    = j >> 2;
  const int rr   = j & 3;
  const float* __restrict__ Vrow = pool + (size_t)n * DWA_D + DWA_UEND + rr * DWA_DA;

  // Warm WGP$ for this lane's B stream (1 KB row -> 4 lines).
  __builtin_prefetch(Vrow,       0, 0);
  __builtin_prefetch(Vrow + 64,  0, 0);
  __builtin_prefetch(Vrow + 128, 0, 0);
  __builtin_prefetch(Vrow + 192, 0, 0);

  wait_async0();
  __syncthreads();

  v8f acc = {};
  #pragma unroll 8
  for (int k0 = 0; k0 < DWA_DA; k0 += 4) {
    v2f a = *reinterpret_cast<const v2f*>(&As[ln][k0 + kh]);
    v2f b = *reinterpret_cast<const v2f*>(Vrow + k0 + kh);
    acc = wmma_f32(a, b, acc);
  }

  const int mbase = (lane >> 4) << 3;  // C/D: lanes 16-31 hold M=8..15
  #pragma unroll
  for (int v = 0; v < 8; ++v) {
    const int brow = rowbase + mbase + v;
    const float sc = alpha[(size_t)brow * DWA_N + n];
    S[(size_t)brow * DWA_NR + j] = acc[v] * sc;
  }
}

// Kernel 2: per 16-row block, accumulate
//   h_A @ W_base^T  (K=256)  +  alpha @ bias (K=512)  +  S @ Umat (K=2048)
// then y = h_A + gamma*(acc + b_base), LayerNorm(256), scale/bias -> out.
// 8 waves, each wave owns two 16x16 column tiles (256 cols total).
__global__ __launch_bounds__(256) void k_asm_ln(
    const float* __restrict__ hA, const float* __restrict__ pool,
    const float* __restrict__ alpha, const float* __restrict__ Wb,
    const float* __restrict__ bb, const float* __restrict__ gammap,
    const float* __restrict__ lnS, const float* __restrict__ lnB,
    const float* __restrict__ S, float* __restrict__ out)
{
  __shared__ float As[16][260];   // h_A tile (also residual source)
  __shared__ float Al[16][516];   // alpha tile
  __shared__ float Y[16][260];    // pre-LN result
  __shared__ float red[2][16][16];

  const int rowbase = blockIdx.x * 16;

  // Async-stage h_A (16x256) and alpha (16x512) tiles into LDS.
  #pragma unroll
  for (int i = threadIdx.x; i < 16 * 64; i += 256) {
    const int r  = i >> 6;
    const int c4 = (i & 63) << 2;
    async_copy_b128(&As[r][c4], hA + (size_t)(rowbase + r) * DWA_DA + c4);
  }
  #pragma unroll
  for (int i = threadIdx.x; i < 16 * 128; i += 256) {
    const int r  = i >> 7;
    const int c4 = (i & 127) << 2;
    async_copy_b128(&Al[r][c4], alpha + (size_t)(rowbase + r) * DWA_N + c4);
  }
  wait_async0();
  __syncthreads();

  const int wave = threadIdx.x >> 5;
  const int lane = threadIdx.x & 31;
  const int ln   = lane & 15;
  const int kh   = (lane >> 4) << 1;
  const float g  = gammap[0];

  for (int t = 0; t < 2; ++t) {
    const int c = (wave + t * 8) * 16 + ln;
    v8f acc = {};

    // (1) base term: B[k=a][c] = W_base[c, a] (rows contiguous)
    const float* __restrict__ Wrow = Wb + (size_t)c * DWA_DA;
    __builtin_prefetch(Wrow,       0, 0);
    __builtin_prefetch(Wrow + 128, 0, 0);
    #pragma unroll 8
    for (int k0 = 0; k0 < DWA_DA; k0 += 4) {
      v2f a = *reinterpret_cast<const v2f*>(&As[ln][k0 + kh]);
      v2f b = *reinterpret_cast<const v2f*>(Wrow + k0 + kh);
      acc = wmma_f32(a, b, acc);
    }

    // (2) bias term: B[k=n][c] = pool[n*4096 + 2048 + c]
    #pragma unroll 4
    for (int k0 = 0; k0 < DWA_N; k0 += 4) {
      v2f a = *reinterpret_cast<const v2f*>(&Al[ln][k0 + kh]);
      v2f b;
      b.x = pool[(size_t)(k0 + kh) * DWA_D + DWA_VEND + c];
      b.y = pool[(size_t)(k0 + kh + 1) * DWA_D + DWA_VEND + c];
      acc = wmma_f32(a, b, acc);
    }

    // (3) main term: A = S rows, B[k=j][c] = pool[(j>>2)*4096 + c*4 + (j&3)]
    //     k0 is a multiple of 4 -> the 4 K-values of one step share n = k0>>2,
    //     and the per-lane pair (kh, kh+1) is contiguous in memory.
    const float* __restrict__ Srow = S + (size_t)(rowbase + ln) * DWA_NR;
    #pragma unroll 8
    for (int k0 = 0; k0 < DWA_NR; k0 += 4) {
      v2f a = *reinterpret_cast<const v2f*>(Srow + k0 + kh);
      const int n = k0 >> 2;
      v2f b = *reinterpret_cast<const v2f*>(pool + (size_t)n * DWA_D + c * DWA_R + kh);
      acc = wmma_f32(a, b, acc);
    }

    // Epilogue: residual + gamma scale into LDS for LayerNorm
    const float bbc = bb[c];
    const int mbase = (lane >> 4) << 3;
    #pragma unroll
    for (int v = 0; v < 8; ++v) {
      const int m = mbase + v;
      Y[m][c] = As[m][c] + g * (acc[v] + bbc);
    }
  }
  __syncthreads();

  // LayerNorm over 256 columns: 16 threads per row, 16 elements each.
  const int row = threadIdx.x >> 4;
  const int sub = threadIdx.x & 15;
  float s = 0.f, ss = 0.f;
  #pragma unroll
  for (int i = 0; i < 16; ++i) {
    const float v = Y[row][sub * 16 + i];
    s += v; ss += v * v;
  }
  red[0][row][sub] = s; red[1][row][sub] = ss;
  __syncthreads();
  float mean = 0.f, m2 = 0.f;
  #pragma unroll
  for (int i = 0; i < 16; ++i) { mean += red[0][row][i]; m2 += red[1][row][i]; }
  mean *= (1.f / DWA_DA);
  const float var = m2 * (1.f / DWA_DA) - mean * mean;
  const float inv = rsqrtf(var + 1e-5f);
  #pragma unroll 4
  for (int i = 0; i < 16; ++i) {
    const int c = sub * 16 + i;
    out[(size_t)(rowbase + row) * DWA_DA + c] =
        (Y[row][c] - mean) * inv * lnS[c] + lnB[c];
  }
}

extern "C" void kernel_launch(void* const* d_in, const int* in_sizes, int n_in,
                              void* d_out, int out_size, void* d_ws, size_t ws_size,
                              hipStream_t stream) {
  const float* hA    = (const float*)d_in[0];
  const float* pool  = (const float*)d_in[1];
  const float* alpha = (const float*)d_in[2];
  const float* Wb    = (const float*)d_in[3];
  const float* bb    = (const float*)d_in[4];
  const float* gamma = (const float*)d_in[5];
  const float* lnS   = (const float*)d_in[6];
  const float* lnB   = (const float*)d_in[7];
  float* S   = (float*)d_ws;   // 2048 x 2048 fp32 = 16 MB scratch
  float* out = (float*)d_out;

  dim3 g1(DWA_B / 16, DWA_NR / 128);
  k_tv<<<g1, 256, 0, stream>>>(hA, pool, alpha, S);
  k_asm_ln<<<DWA_B / 16, 256, 0, stream>>>(hA, pool, alpha, Wb, bb, gamma,
                                           lnS, lnB, S, out);
}